// SLayerRgbNN_53927609368896
// MI455X (gfx1250) — compile-verified
//
#include <hip/hip_runtime.h>
#include <hip/hip_bf16.h>

#define NV   32    // views
#define NB   128   // batch
#define NP   256   // points
#define NJ   75    // elements
#define NFLT 32    // n_filters
#define NS2  25    // stage_2 out
#define K1   (NV * NS2)   // 800
#define N1   500
#define N2   200
#define BN_EPS 1e-5f

typedef __attribute__((ext_vector_type(2))) float v2f;
typedef __attribute__((ext_vector_type(8))) float v8f;

// ---------------------------------------------------------------------------
// Kernel 1: transform + SLayer.  One block per (v,b); points staged in LDS,
// then 75 threads (one per structure element j) sweep all 256 points.
// ---------------------------------------------------------------------------
__global__ void slayer_kernel(const float* __restrict__ births,
                              const float* __restrict__ lifetimes,
                              const int*   __restrict__ mask,
                              const float* __restrict__ centers,
                              const float* __restrict__ sharpness,
                              float* __restrict__ z) {
  __shared__ float qx[NP], qy[NP], qm[NP];
  const int vb = blockIdx.x;
  const int v  = vb / NB;
  const size_t base = (size_t)vb * NP;
  const float inv = 0.70710678118654752440f; // 1/sqrt(2)
  for (int p = threadIdx.x; p < NP; p += blockDim.x) {
    float bb = births[base + p];
    float lf = lifetimes[base + p];
    float dd = bb + lf + 0.01f;
    float x = (bb + dd) * inv;
    float y = (dd - bb) * inv;                 // always > 0
    if (y <= 0.1f) y = __logf(y * 10.0f) * 0.1f + 0.1f;
    qx[p] = x; qy[p] = y;
    qm[p] = (float)mask[base + p];
  }
  __syncthreads();
  for (int j = threadIdx.x; j < NJ; j += blockDim.x) {
    const size_t cj = ((size_t)v * NJ + j) * 2;
    float cx = centers[cj + 0], cy = centers[cj + 1];
    float sx = sharpness[cj + 0], sy = sharpness[cj + 1];
    float s2x = sx * sx, s2y = sy * sy;
    float acc = 0.0f;
    #pragma unroll 4
    for (int p = 0; p < NP; ++p) {
      float dx = qx[p] - cx;
      float dy = qy[p] - cy;
      float d2 = s2x * dx * dx + s2y * dy * dy;
      acc = __builtin_fmaf(qm[p], __expf(-d2), acc);
    }
    z[(size_t)vb * NJ + j] = acc;
  }
}

// ---------------------------------------------------------------------------
// Kernel 2: collapse the two 1x1 convs: Weff[v,g,c] = sum_f w2[v,g,f]*w1[v,f,c]
// ---------------------------------------------------------------------------
__global__ void fuse_w_kernel(const float* __restrict__ w1,
                              const float* __restrict__ w2,
                              float* __restrict__ Weff) {
  int idx = blockIdx.x * blockDim.x + threadIdx.x;
  if (idx >= NV * 8 * 3) return;
  int v = idx / 24, r = idx % 24, g = r / 3, c = r % 3;
  float acc = 0.0f;
  for (int f = 0; f < NFLT; ++f)
    acc += w2[((size_t)v * 8 + g) * NFLT + f] * w1[((size_t)v * NFLT + f) * 3 + c];
  Weff[idx] = acc;
}

// ---------------------------------------------------------------------------
// Kernel 3: stage_1 (neighbor-view mix + channel max)
// ---------------------------------------------------------------------------
__global__ void stage1_kernel(const float* __restrict__ z,
                              const float* __restrict__ Weff,
                              float* __restrict__ h) {
  int idx = blockIdx.x * blockDim.x + threadIdx.x;
  if (idx >= NV * NB * NJ) return;
  int v = idx / (NB * NJ);
  int rem = idx % (NB * NJ);
  int b = rem / NJ, j = rem % NJ;
  int vm = (v + NV - 1) % NV, vp = (v + 1) % NV;
  float zm = z[((size_t)vm * NB + b) * NJ + j];
  float z0 = z[((size_t)v  * NB + b) * NJ + j];
  float zp = z[((size_t)vp * NB + b) * NJ + j];
  const float* Wv = Weff + v * 24;
  float best = -1e30f;
  #pragma unroll
  for (int g = 0; g < 8; ++g) {
    float t = Wv[g * 3 + 0] * zm + Wv[g * 3 + 1] * z0 + Wv[g * 3 + 2] * zp;
    best = fmaxf(best, t);
  }
  h[idx] = best;
}

// ---------------------------------------------------------------------------
// Kernel 4: stage_2 linear 1 (J -> S2), per view
// ---------------------------------------------------------------------------
__global__ void stage2_l1_kernel(const float* __restrict__ h,
                                 const float* __restrict__ l1w,
                                 const float* __restrict__ l1b,
                                 float* __restrict__ u1) {
  int idx = blockIdx.x * blockDim.x + threadIdx.x;
  if (idx >= NV * NB * NS2) return;
  int v = idx / (NB * NS2);
  int rem = idx % (NB * NS2);
  int b = rem / NS2, o = rem % NS2;
  const float* hr = h + ((size_t)v * NB + b) * NJ;
  const float* wr = l1w + ((size_t)v * NS2 + o) * NJ;
  float acc = l1b[(size_t)v * NS2 + o];
  #pragma unroll 5
  for (int j = 0; j < NJ; ++j) acc = __builtin_fmaf(hr[j], wr[j], acc);
  u1[idx] = acc;
}

// ---------------------------------------------------------------------------
// Kernel 5: training-mode BatchNorm over batch axis, per (v,o) column
// ---------------------------------------------------------------------------
__global__ void bn1_kernel(float* __restrict__ u1,
                           const float* __restrict__ g,
                           const float* __restrict__ bt) {
  int col = blockIdx.x * blockDim.x + threadIdx.x; // col = v*NS2 + o
  if (col >= NV * NS2) return;
  int v = col / NS2, o = col % NS2;
  float mean = 0.f, m2 = 0.f;
  for (int b = 0; b < NB; ++b) {
    float x = u1[((size_t)v * NB + b) * NS2 + o];
    mean += x; m2 += x * x;
  }
  mean *= (1.0f / NB);
  float var = m2 * (1.0f / NB) - mean * mean;  // biased (training BN)
  float scl = rsqrtf(var + BN_EPS) * g[col];
  float beta = bt[col];
  for (int b = 0; b < NB; ++b) {
    size_t i = ((size_t)v * NB + b) * NS2 + o;
    u1[i] = (u1[i] - mean) * scl + beta;
  }
}

// ---------------------------------------------------------------------------
// Kernel 6: stage_2 linear 2 (S2 -> S2) + ReLU, scattered into X[B, V*S2]
// ---------------------------------------------------------------------------
__global__ void stage2_l2_kernel(const float* __restrict__ u1,
                                 const float* __restrict__ l2w,
                                 const float* __restrict__ l2b,
                                 float* __restrict__ X) {
  int idx = blockIdx.x * blockDim.x + threadIdx.x;
  if (idx >= NV * NB * NS2) return;
  int v = idx / (NB * NS2);
  int rem = idx % (NB * NS2);
  int b = rem / NS2, p = rem % NS2;
  const float* ur = u1 + ((size_t)v * NB + b) * NS2;
  const float* wr = l2w + ((size_t)v * NS2 + p) * NS2;
  float acc = l2b[(size_t)v * NS2 + p];
  #pragma unroll
  for (int o = 0; o < NS2; ++o) acc = __builtin_fmaf(ur[o], wr[o], acc);
  acc = fmaxf(acc, 0.0f);
  X[(size_t)b * K1 + (size_t)v * NS2 + p] = acc;
}

// ---------------------------------------------------------------------------
// Kernel 7: WMMA fp32 GEMM  Y[M,N] = X[M,K] * W[N,K]^T + bias[N]
// One wave (32 threads) per 16x16 output tile; K stepped by 4 via
// V_WMMA_F32_16X16X4_F32.  EXEC stays all-ones throughout.
//
// N-edge handling: out-of-range lanes read W row (N-1) (in-bounds address).
// In D = A*B, output column n depends only on B column n, so the garbage in
// padded B columns can only reach padded D columns, which the store guard
// discards.  No per-iteration masking needed -> inner loop is pure load+WMMA.
//
// A frag (16x4 f32): lane m = lane&15; vgpr{0,1} = K = (lane>>4)*2 + {0,1}
// B frag (4x16 f32): lane n = lane&15; vgpr{0,1} = K = (lane>>4)*2 + {0,1}
// C/D (16x16 f32):   lane n = lane&15; vgpr i -> row (lane>>4)*8 + i
// ---------------------------------------------------------------------------
__global__ void wmma_fc_kernel(const float* __restrict__ X,
                               const float* __restrict__ W,
                               const float* __restrict__ bias,
                               float* __restrict__ Y,
                               int M, int K, int N) {
  const int lane = threadIdx.x;       // 0..31, one full wave per block
  const int half = lane >> 4;         // K sub-pair selector
  const int l15  = lane & 15;
  const int m = blockIdx.y * 16 + l15;     // A row  (M divisible by 16 here)
  const int n = blockIdx.x * 16 + l15;     // B col / D col
  const int nc = (n < N) ? n : (N - 1);    // clamped in-bounds address

  const float* xrow = X + (size_t)m  * K + half * 2;
  const float* wrow = W + (size_t)nc * K + half * 2;

  v8f acc = {};
  #pragma unroll 4
  for (int k0 = 0; k0 < K; k0 += 4) {
    v2f a = *(const v2f*)(xrow + k0);
    v2f b = *(const v2f*)(wrow + k0);
    acc = __builtin_amdgcn_wmma_f32_16x16x4_f32(
        /*neg_a=*/false, a, /*neg_b=*/false, b,
        /*c_mod=*/(short)0, acc, /*reuse_a=*/false, /*reuse_b=*/false);
  }

  if (n < N) {
    const int rbase = blockIdx.y * 16 + half * 8;
    const float bv = bias[n];
    #pragma unroll
    for (int i = 0; i < 8; ++i)
      Y[(size_t)(rbase + i) * N + n] = acc[i] + bv;
  }
}

// ---------------------------------------------------------------------------
// Kernel 8: BatchNorm over batch for Y[NB, N1]
// ---------------------------------------------------------------------------
__global__ void bn2_kernel(float* __restrict__ Y,
                           const float* __restrict__ g,
                           const float* __restrict__ bt) {
  int col = blockIdx.x * blockDim.x + threadIdx.x;
  if (col >= N1) return;
  float mean = 0.f, m2 = 0.f;
  for (int b = 0; b < NB; ++b) {
    float x = Y[(size_t)b * N1 + col];
    mean += x; m2 += x * x;
  }
  mean *= (1.0f / NB);
  float var = m2 * (1.0f / NB) - mean * mean;
  float scl = rsqrtf(var + BN_EPS) * g[col];
  float beta = bt[col];
  for (int b = 0; b < NB; ++b) {
    size_t i = (size_t)b * N1 + col;
    Y[i] = (Y[i] - mean) * scl + beta;
  }
}

// ---------------------------------------------------------------------------
extern "C" void kernel_launch(void* const* d_in, const int* in_sizes, int n_in,
                              void* d_out, int out_size, void* d_ws, size_t ws_size,
                              hipStream_t stream) {
  const float* births    = (const float*)d_in[0];
  const float* lifetimes = (const float*)d_in[1];
  const int*   mask      = (const int*)  d_in[2];
  const float* centers   = (const float*)d_in[3];
  const float* sharpness = (const float*)d_in[4];
  const float* w1        = (const float*)d_in[5];
  const float* w2        = (const float*)d_in[6];
  const float* l1_w      = (const float*)d_in[7];
  const float* l1_b      = (const float*)d_in[8];
  const float* bn1_g     = (const float*)d_in[9];
  const float* bn1_b     = (const float*)d_in[10];
  const float* l2_w      = (const float*)d_in[11];
  const float* l2_b      = (const float*)d_in[12];
  const float* fc1_w     = (const float*)d_in[13];
  const float* fc1_b     = (const float*)d_in[14];
  const float* bn2_g     = (const float*)d_in[15];
  const float* bn2_b     = (const float*)d_in[16];
  const float* fc2_w     = (const float*)d_in[17];
  const float* fc2_b     = (const float*)d_in[18];
  float* out = (float*)d_out;

  // workspace layout (floats)
  float* ws   = (float*)d_ws;
  float* z    = ws;                        // [V,B,J]   307200
  float* h    = z    + NV * NB * NJ;       // [V,B,J]   307200
  float* Weff = h    + NV * NB * NJ;       // [V,8,3]      768
  float* u1   = Weff + NV * 8 * 3;         // [V,B,S2]  102400
  float* X    = u1   + NV * NB * NS2;      // [B,800]   102400
  float* Y    = X    + NB * K1;            // [B,500]    64000

  slayer_kernel<<<NV * NB, 128, 0, stream>>>(births, lifetimes, mask,
                                             centers, sharpness, z);

  fuse_w_kernel<<<(NV * 8 * 3 + 127) / 128, 128, 0, stream>>>(w1, w2, Weff);

  stage1_kernel<<<(NV * NB * NJ + 255) / 256, 256, 0, stream>>>(z, Weff, h);

  stage2_l1_kernel<<<(NV * NB * NS2 + 255) / 256, 256, 0, stream>>>(h, l1_w, l1_b, u1);

  bn1_kernel<<<(NV * NS2 + 127) / 128, 128, 0, stream>>>(u1, bn1_g, bn1_b);

  stage2_l2_kernel<<<(NV * NB * NS2 + 255) / 256, 256, 0, stream>>>(u1, l2_w, l2_b, X);

  // fc1: [128,800] x [500,800]^T -> Y[128,500]
  {
    dim3 grid((N1 + 15) / 16, NB / 16);
    wmma_fc_kernel<<<grid, 32, 0, stream>>>(X, fc1_w, fc1_b, Y, NB, K1, N1);
  }

  bn2_kernel<<<(N1 + 127) / 128, 128, 0, stream>>>(Y, bn2_g, bn2_b);

  // fc2: [128,500] x [200,500]^T -> out[128,200]
  {
    dim3 grid((N2 + 15) / 16, NB / 16);
    wmma_fc_kernel<<<grid, 32, 0, stream>>>(Y, fc2_w, fc2_b, out, NB, N1, N2);
  }
}